// BilinearInterpolate_29085518528596
// MI455X (gfx1250) — compile-verified
//
#include <hip/hip_runtime.h>
#include <hip/hip_bf16.h>

#ifndef __has_builtin
#define __has_builtin(x) 0
#endif

typedef __attribute__((ext_vector_type(4))) float f4;
typedef __attribute__((ext_vector_type(4))) int   v4i;

#define GLOBAL_AS __attribute__((address_space(1)))
#define LDS_AS    __attribute__((address_space(3)))

static constexpr int N  = 8;
static constexpr int H  = 256;
static constexpr int W  = 256;
static constexpr int C  = 32;
static constexpr int OH = 512;
static constexpr int OW = 512;

// -------- async global -> LDS copy (CDNA5 GLOBAL_LOAD_ASYNC_TO_LDS_B128) ----
__device__ __forceinline__ void async_copy_b128(const float* gsrc, void* lds_dst) {
#if __has_builtin(__builtin_amdgcn_global_load_async_to_lds_b128)
    __builtin_amdgcn_global_load_async_to_lds_b128(
        (GLOBAL_AS v4i*)gsrc, (LDS_AS v4i*)lds_dst, /*offset=*/0, /*cpol=*/0);
#else
    unsigned lds_off = (unsigned)(unsigned long)(LDS_AS void*)lds_dst;
    asm volatile("global_load_async_to_lds_b128 %0, %1, off"
                 :: "v"(lds_off), "v"(gsrc)
                 : "memory");
#endif
}

__device__ __forceinline__ void wait_async_zero() {
#if __has_builtin(__builtin_amdgcn_s_wait_asynccnt)
    __builtin_amdgcn_s_wait_asynccnt(0);
#else
    asm volatile("s_wait_asynccnt 0" ::: "memory");
#endif
}

// One block per (n, r): stages input rows r and r+1 in LDS via the async
// tensor path, produces output rows 2r+1, 2r+2 (plus row 0 for r==0).
__global__ __launch_bounds__(256)
void BilinearInterpolate_29085518528596_kernel(const float* __restrict__ img,
                                               float* __restrict__ out) {
    __shared__ f4 lds[2 * (W * C / 4)];           // 2 rows * 32KB = 64KB

    const int tid = threadIdx.x;                  // 0..255
    const int b   = blockIdx.x;
    const int n   = b >> 8;                       // image index
    const int r   = b & 255;                      // input row pair base
    const int r2  = r + (r < H - 1);              // clamp (matches reference)

    const float* row0 = img + (size_t)(n * H + r ) * (W * C);
    const float* row1 = img + (size_t)(n * H + r2) * (W * C);

    // ---- Stage: 2 rows x 2048 float4 each, 16 async B128 copies per thread.
    const int ROWQ = W * C / 4;                   // 2048 float4 per input row
    #pragma unroll
    for (int i = 0; i < ROWQ / 256; ++i) {
        int q = tid + (i << 8);
        async_copy_b128(row0 + q * 4, &lds[q]);
        async_copy_b128(row1 + q * 4, &lds[ROWQ + q]);
    }
    wait_async_zero();
    __syncthreads();

    // ---- Which output rows this block owns (top-weight, bottom-weight).
    int   oys[3]; float wts[3], wbs[3];
    int nt = 0;
    if (r == 0) { oys[nt] = 0;         wts[nt] = 1.00f; wbs[nt] = 0.00f; ++nt; }
    {             oys[nt] = 2 * r + 1; wts[nt] = 0.75f; wbs[nt] = 0.25f; ++nt; }
    if (2 * r + 2 < OH) {
                  oys[nt] = 2 * r + 2; wts[nt] = 0.25f; wbs[nt] = 0.75f; ++nt; }

    // ---- Compute: each output row is 4096 float4; 16 per thread, coalesced.
    for (int t = 0; t < nt; ++t) {
        f4* __restrict__ outrow =
            (f4*)(out + (size_t)(n * OH + oys[t]) * (OW * C));
        const float wt = wts[t], wb = wbs[t];

        #pragma unroll 4
        for (int k = 0; k < (OW * C / 4) / 256; ++k) {
            const int idx = tid + (k << 8);       // float4 index in output row
            const int ox  = idx >> 3;             // output pixel (8 f4 per px)
            const int cg  = idx & 7;              // channel group of 4

            float gx = fmaxf((ox + 0.5f) * 0.5f - 0.5f, 0.0f);
            int   x0 = (int)gx;                   // floor, gx >= 0
            float w0 = gx - (float)x0;
            float w1 = 1.0f - w0;
            int   x1 = x0 + (x0 < W - 1);

            f4 t0 = lds[x0 * 8 + cg];
            f4 t1 = lds[x1 * 8 + cg];
            f4 b0 = lds[ROWQ + x0 * 8 + cg];
            f4 b1 = lds[ROWQ + x1 * 8 + cg];

            // Same associativity as reference: h1*(w1*v00+w0*v01)+h0*(...)
            f4 o = wt * (w1 * t0 + w0 * t1) + wb * (w1 * b0 + w0 * b1);

            // Streaming store: output is write-once, keep L2 for input rows.
            __builtin_nontemporal_store(o, &outrow[idx]);
        }
    }
}

extern "C" void kernel_launch(void* const* d_in, const int* in_sizes, int n_in,
                              void* d_out, int out_size, void* d_ws, size_t ws_size,
                              hipStream_t stream) {
    (void)in_sizes; (void)n_in; (void)out_size; (void)d_ws; (void)ws_size;
    const float* img = (const float*)d_in[0];
    float*       out = (float*)d_out;
    dim3 grid(N * H);        // 2048 blocks: one per (n, input-row r)
    dim3 block(256);         // 8 wave32 waves
    BilinearInterpolate_29085518528596_kernel<<<grid, block, 0, stream>>>(img, out);
}